// GravNetLayer_30494267802109
// MI455X (gfx1250) — compile-verified
//
#include <hip/hip_runtime.h>
#include <math.h>

typedef float v2f __attribute__((ext_vector_type(2)));
typedef float v8f __attribute__((ext_vector_type(8)));

#define BATCH 4
#define NPTS 4096
#define FEAT 32
#define NS 4
#define NLR 22
#define CO 26          // NS + NLR
#define KNN 40
#define SLR_STRIDE 32  // padded 26 -> 32
#define FP_DIM 108     // 64 + 22 + 22
#define FP_STRIDE 112  // padded
#define OUT_DIM 48

__device__ __forceinline__ v8f wmma_f32_16x16x4(v2f a, v2f b, v8f c) {
  // D = A(16x4, fp32) x B(4x16, fp32) + C(16x16, fp32), exact fp32 accumulate
  return __builtin_amdgcn_wmma_f32_16x16x4_f32(false, a, false, b, (short)0, c,
                                               false, false);
}

// ---------------- kernel 1: per-batch feature means over N ----------------
__global__ void k_mean(const float* __restrict__ x, float* __restrict__ mean) {
  __shared__ float red[256];
  int b = blockIdx.x;
  int t = threadIdx.x;
  int f = t & 31;       // feature
  int i = t >> 5;       // 0..7 partial id
  const float* xb = x + (size_t)b * NPTS * FEAT;
  float s = 0.f;
  for (int n = i; n < NPTS; n += 8) s += xb[n * FEAT + f];
  red[t] = s;
  __syncthreads();
  if (i == 0) {
    float tot = 0.f;
    for (int j = 0; j < 8; ++j) tot += red[j * 32 + f];
    mean[b * FEAT + f] = tot * (1.0f / NPTS);
  }
}

// ------------- kernel 2: slr = relu(xm @ W_slr + b), WMMA fp32 -------------
// one wave (32 threads) per 16-row tile; cols 0..25 valid, 26..31 zero-padded.
// Out-of-range weight columns are fetched from a clamped (valid) address and
// zeroed with a data-select, so no EXEC divergence in the WMMA loop.
__global__ void k_slr(const float* __restrict__ x, const float* __restrict__ mean,
                      const float* __restrict__ W, const float* __restrict__ bias,
                      float* __restrict__ slr) {
  int tile = blockIdx.x;            // 1024 tiles of 16 rows
  int lane = threadIdx.x;           // 0..31
  int r = lane & 15;
  int kl = (lane >> 4) << 1;        // 0 or 2
  int rowA = tile * 16 + r;         // global row for A fragment
  int b = rowA >> 12;
  int n = rowA & (NPTS - 1);
  const float* xr = x + ((size_t)b * NPTS + n) * FEAT;
  const float* mb = mean + b * FEAT;
  int c_lo = r;
  int c_hi = 16 + r;
  bool hi_valid = (c_hi < CO);
  int c_hi_clamp = hi_valid ? c_hi : (CO - 1);  // always a valid address
  float hi_mask = hi_valid ? 1.0f : 0.0f;
  v8f c0 = {0.f, 0.f, 0.f, 0.f, 0.f, 0.f, 0.f, 0.f};
  v8f c1 = c0;
  // first half of K: A comes from x
  for (int k0 = 0; k0 < FEAT; k0 += 4) {
    int ka = k0 + kl;
    v2f a;
    a.x = xr[ka];
    a.y = xr[ka + 1];
    v2f b0, b1;
    b0.x = W[ka * CO + c_lo];
    b0.y = W[(ka + 1) * CO + c_lo];
    b1.x = W[ka * CO + c_hi_clamp] * hi_mask;
    b1.y = W[(ka + 1) * CO + c_hi_clamp] * hi_mask;
    c0 = wmma_f32_16x16x4(a, b0, c0);
    c1 = wmma_f32_16x16x4(a, b1, c1);
  }
  // second half of K: A comes from the broadcast mean
  for (int k0 = FEAT; k0 < 2 * FEAT; k0 += 4) {
    int ka = k0 + kl;
    v2f a;
    a.x = mb[ka - FEAT];
    a.y = mb[ka + 1 - FEAT];
    v2f b0, b1;
    b0.x = W[ka * CO + c_lo];
    b0.y = W[(ka + 1) * CO + c_lo];
    b1.x = W[ka * CO + c_hi_clamp] * hi_mask;
    b1.y = W[(ka + 1) * CO + c_hi_clamp] * hi_mask;
    c0 = wmma_f32_16x16x4(a, b0, c0);
    c1 = wmma_f32_16x16x4(a, b1, c1);
  }
  int rbase = tile * 16 + ((lane >> 4) << 3);  // +0 or +8
  float bias_lo = bias[c_lo];
  float bias_hi = bias[c_hi_clamp] * hi_mask;
#pragma unroll
  for (int v = 0; v < 8; ++v) {
    size_t row = rbase + v;
    float v0 = fmaxf(c0[v] + bias_lo, 0.f);
    float v1 = fmaxf(c1[v] * hi_mask + bias_hi, 0.f) * hi_mask;
    slr[row * SLR_STRIDE + c_lo] = v0;
    slr[row * SLR_STRIDE + c_hi] = v1;
  }
}

// ------ kernel 3: kNN (top-40 by distance in 4-d s space) + pooling -------
// 128 rows per block (all in same batch); candidates staged through LDS;
// per-row sorted top-40 list in LDS (strict > keeps jax top_k tie order).
__global__ void k_knn(const float* __restrict__ x, const float* __restrict__ mean,
                      const float* __restrict__ slr, float* __restrict__ fp) {
  __shared__ float sh_s[128 * 4];
  __shared__ float topd[KNN * 128];
  __shared__ int   topi[KNN * 128];
  int t = threadIdx.x;                 // 0..127
  int row = blockIdx.x * 128 + t;      // 0..16383
  int b = row >> 12;
  int n = row & (NPTS - 1);
  const float* srow = slr + (size_t)row * SLR_STRIDE;
  float s0 = srow[0], s1 = srow[1], s2 = srow[2], s3 = srow[3];
  const float* sb = slr + (size_t)b * NPTS * SLR_STRIDE;
  int cnt = 0;
  for (int m0 = 0; m0 < NPTS; m0 += 128) {
    __syncthreads();
    {
      const float* src = sb + (size_t)(m0 + t) * SLR_STRIDE;
      sh_s[t * 4 + 0] = src[0];
      sh_s[t * 4 + 1] = src[1];
      sh_s[t * 4 + 2] = src[2];
      sh_s[t * 4 + 3] = src[3];
    }
    __syncthreads();
    for (int mm = 0; mm < 128; ++mm) {
      float e0 = s0 - sh_s[mm * 4 + 0];
      float e1 = s1 - sh_s[mm * 4 + 1];
      float e2 = s2 - sh_s[mm * 4 + 2];
      float e3 = s3 - sh_s[mm * 4 + 3];
      float d2 = e0 * e0 + e1 * e1 + e2 * e2 + e3 * e3;
      float d = sqrtf(fmaxf(d2, 0.f));
      if (cnt < KNN) {
        int j = cnt++;
        while (j > 0 && topd[(j - 1) * 128 + t] > d) {
          topd[j * 128 + t] = topd[(j - 1) * 128 + t];
          topi[j * 128 + t] = topi[(j - 1) * 128 + t];
          --j;
        }
        topd[j * 128 + t] = d;
        topi[j * 128 + t] = m0 + mm;
      } else if (d < topd[(KNN - 1) * 128 + t]) {
        int j = KNN - 1;
        while (j > 0 && topd[(j - 1) * 128 + t] > d) {
          topd[j * 128 + t] = topd[(j - 1) * 128 + t];
          topi[j * 128 + t] = topi[(j - 1) * 128 + t];
          --j;
        }
        topd[j * 128 + t] = d;
        topi[j * 128 + t] = m0 + mm;
      }
    }
  }
  // neighbor feature pooling: ft = lr[idx] * exp(-10 d^2); mean + max over K
  float fmean[NLR];
  float fmx[NLR];
#pragma unroll
  for (int j = 0; j < NLR; ++j) { fmean[j] = 0.f; fmx[j] = -__builtin_inff(); }
  for (int k = 0; k < KNN; ++k) {
    float d = topd[k * 128 + t];
    int idx = topi[k * 128 + t];
    float v = expf(-10.0f * d * d);
    const float* lrp = sb + (size_t)idx * SLR_STRIDE + NS;
#pragma unroll
    for (int j = 0; j < NLR; ++j) {
      float f = lrp[j] * v;
      fmean[j] += f;
      fmx[j] = fmaxf(fmx[j], f);
    }
  }
  float* fpr = fp + (size_t)row * FP_STRIDE;
  const float* xr = x + ((size_t)b * NPTS + n) * FEAT;
  const float* mb = mean + b * FEAT;
  for (int f = 0; f < FEAT; ++f) {
    fpr[f] = xr[f];
    fpr[FEAT + f] = mb[f];
  }
#pragma unroll
  for (int j = 0; j < NLR; ++j) {
    fpr[2 * FEAT + j] = fmean[j] * (1.0f / KNN);
    fpr[2 * FEAT + NLR + j] = fmx[j];
  }
  fpr[108] = 0.f; fpr[109] = 0.f; fpr[110] = 0.f; fpr[111] = 0.f;  // pad
}

// ------------- kernel 4: out = relu(fp @ W_out + b), WMMA fp32 -------------
__global__ void k_out(const float* __restrict__ fp, const float* __restrict__ W,
                      const float* __restrict__ bias, float* __restrict__ out) {
  int tile = blockIdx.x;
  int lane = threadIdx.x;
  int r = lane & 15;
  int kl = (lane >> 4) << 1;
  int rowA = tile * 16 + kl;  // note: A row index is lane&15, handled below
  rowA = tile * 16 + r;
  const float* ar = fp + (size_t)rowA * FP_STRIDE;
  v8f c0 = {0.f, 0.f, 0.f, 0.f, 0.f, 0.f, 0.f, 0.f};
  v8f c1 = c0, c2 = c0;
  for (int k0 = 0; k0 < FP_DIM; k0 += 4) {  // 108 = 27 * 4
    int ka = k0 + kl;
    v2f a;
    a.x = ar[ka];
    a.y = ar[ka + 1];
    v2f b0, b1, b2;
    b0.x = W[ka * OUT_DIM + r];       b0.y = W[(ka + 1) * OUT_DIM + r];
    b1.x = W[ka * OUT_DIM + 16 + r];  b1.y = W[(ka + 1) * OUT_DIM + 16 + r];
    b2.x = W[ka * OUT_DIM + 32 + r];  b2.y = W[(ka + 1) * OUT_DIM + 32 + r];
    c0 = wmma_f32_16x16x4(a, b0, c0);
    c1 = wmma_f32_16x16x4(a, b1, c1);
    c2 = wmma_f32_16x16x4(a, b2, c2);
  }
  int rbase = tile * 16 + ((lane >> 4) << 3);
  float bl0 = bias[r], bl1 = bias[16 + r], bl2 = bias[32 + r];
#pragma unroll
  for (int v = 0; v < 8; ++v) {
    size_t row = rbase + v;
    out[row * OUT_DIM + r]      = fmaxf(c0[v] + bl0, 0.f);
    out[row * OUT_DIM + 16 + r] = fmaxf(c1[v] + bl1, 0.f);
    out[row * OUT_DIM + 32 + r] = fmaxf(c2[v] + bl2, 0.f);
  }
}

extern "C" void kernel_launch(void* const* d_in, const int* in_sizes, int n_in,
                              void* d_out, int out_size, void* d_ws, size_t ws_size,
                              hipStream_t stream) {
  (void)in_sizes; (void)n_in; (void)out_size; (void)ws_size;
  const float* x     = (const float*)d_in[0];
  const float* W_slr = (const float*)d_in[1];
  const float* b_slr = (const float*)d_in[2];
  const float* W_out = (const float*)d_in[3];
  const float* b_out = (const float*)d_in[4];
  float* out = (float*)d_out;

  float* ws   = (float*)d_ws;
  float* mean = ws;                                        // 128 floats
  float* slr  = ws + 256;                                  // 16384*32 floats
  float* fp   = slr + (size_t)BATCH * NPTS * SLR_STRIDE;   // 16384*112 floats

  k_mean<<<BATCH, 256, 0, stream>>>(x, mean);
  k_slr<<<(BATCH * NPTS) / 16, 32, 0, stream>>>(x, mean, W_slr, b_slr, slr);
  k_knn<<<(BATCH * NPTS) / 128, 128, 0, stream>>>(x, mean, slr, fp);
  k_out<<<(BATCH * NPTS) / 16, 32, 0, stream>>>(fp, W_out, b_out, out);
}